// ChebNetII_60971355734617
// MI455X (gfx1250) — compile-verified
//
#include <hip/hip_runtime.h>
#include <hip/hip_bf16.h>

#define F_IN 256
#define HD   64
#define KCH  6   // Chebyshev order

typedef float v2f __attribute__((ext_vector_type(2)));
typedef float v8f __attribute__((ext_vector_type(8)));

// ---------------------------------------------------------------------------
// coe = 2/(K+1) * T @ relu(temp); T computed like the numpy reference
// (f64 node positions, per-row f32 rounding of the recurrence).
// ---------------------------------------------------------------------------
__global__ void coe_kernel(const float* __restrict__ temp, float* __restrict__ coe) {
    if (threadIdx.x != 0 || blockIdx.x != 0) return;
    const double PI = 3.14159265358979323846;
    double xs[KCH + 1];
    double T[KCH + 1][KCH + 1];
    for (int j = 0; j <= KCH; ++j)
        xs[j] = cos(((double)(KCH - j) + 0.5) * PI / (double)(KCH + 1));
    for (int j = 0; j <= KCH; ++j) {
        T[0][j] = 1.0;
        T[1][j] = (double)(float)xs[j];
    }
    for (int i = 2; i <= KCH; ++i)
        for (int j = 0; j <= KCH; ++j)
            T[i][j] = (double)(float)(2.0 * xs[j] * T[i - 1][j] - T[i - 2][j]);
    for (int i = 0; i <= KCH; ++i) {
        float acc = 0.0f;
        for (int j = 0; j <= KCH; ++j) {
            float t = temp[j];
            t = t > 0.0f ? t : 0.0f;
            acc += (float)T[i][j] * t;
        }
        coe[i] = (2.0f / (float)(KCH + 1)) * acc;
    }
}

// ---------------------------------------------------------------------------
// deg[src[e]] += 1
// ---------------------------------------------------------------------------
__global__ __launch_bounds__(256) void deg_kernel(const int* __restrict__ src,
                                                  float* __restrict__ deg, int E) {
    int i = blockIdx.x * blockDim.x + threadIdx.x;
    if (i < E) atomicAdd(&deg[src[i]], 1.0f);
}

// dis[i] = deg>0 ? rsqrt(max(deg,1)) : 0
__global__ __launch_bounds__(256) void dis_kernel(const float* __restrict__ deg,
                                                  float* __restrict__ dis, int n) {
    int i = blockIdx.x * blockDim.x + threadIdx.x;
    if (i < n) {
        float d = deg[i];
        dis[i] = d > 0.0f ? rsqrtf(fmaxf(d, 1.0f)) : 0.0f;
    }
}

// ---------------------------------------------------------------------------
// Fused MLP: h = relu(X@W1 + b1)@W2 + b2 using V_WMMA_F32_16X16X4_F32.
// Block = 128 threads (4 waves); each block handles a 16-row tile, each wave
// owns a 16-wide column slab of the 64 hidden features.
// LDS row strides padded so A-fragment reads spread across banks (4m mod 64)
// while rows stay 16B aligned (260*4 = 65*16, 68*4 = 17*16).
// ---------------------------------------------------------------------------
#define SXP 260
#define SHP 68

__global__ __launch_bounds__(128) void mlp_kernel(const float* __restrict__ X,
                                                  const float* __restrict__ W1,
                                                  const float* __restrict__ b1,
                                                  const float* __restrict__ W2,
                                                  const float* __restrict__ b2,
                                                  float* __restrict__ h_out, int N) {
    __shared__ float sX[16 * SXP];
    __shared__ float sH1[16 * SHP];

    const int tid  = threadIdx.x;
    const int lane = tid & 31;
    const int wave = tid >> 5;        // 0..3
    const int row0 = blockIdx.x * 16;

    // Cooperative load of 16x256 X tile (clamped rows for safety)
    {
        const int r  = tid >> 3;      // 0..15
        const int c4 = tid & 7;       // 0..7
        const int gr = min(row0 + r, N - 1);
        const float4* srcv = (const float4*)(X + (size_t)gr * F_IN);
        float4* dstv = (float4*)(sX + r * SXP);
#pragma unroll
        for (int i = 0; i < 8; ++i)
            dstv[c4 + 8 * i] = srcv[c4 + 8 * i];
    }
    __syncthreads();

    const int m    = lane & 15;          // M (A) / N (B,C,D) index
    const int half = lane >> 4;          // 0|1
    const int kb   = half * 2;           // K sub-offset within the 4-wide step
    const int col0 = wave * 16;

    // ---- Layer 1: C = Xtile @ W1  (K = 256, 64 WMMA steps) ----
    v8f c = {};
#pragma unroll 8
    for (int k = 0; k < F_IN; k += 4) {
        v2f a, b;
        a.x = sX[m * SXP + k + kb];
        a.y = sX[m * SXP + k + kb + 1];
        b.x = W1[(size_t)(k + kb) * HD + col0 + m];
        b.y = W1[(size_t)(k + kb + 1) * HD + col0 + m];
        c = __builtin_amdgcn_wmma_f32_16x16x4_f32(false, a, false, b,
                                                  (short)0, c, false, false);
    }
    // bias + relu -> LDS
    {
        const float bias1 = b1[col0 + m];
#pragma unroll
        for (int r = 0; r < 8; ++r) {
            float v = c[r] + bias1;
            v = v > 0.0f ? v : 0.0f;
            sH1[(r + 8 * half) * SHP + col0 + m] = v;
        }
    }
    __syncthreads();

    // ---- Layer 2: C2 = H1 @ W2  (K = 64, 16 WMMA steps) ----
    v8f c2 = {};
#pragma unroll
    for (int k = 0; k < HD; k += 4) {
        v2f a, b;
        a.x = sH1[m * SHP + k + kb];
        a.y = sH1[m * SHP + k + kb + 1];
        b.x = W2[(size_t)(k + kb) * HD + col0 + m];
        b.y = W2[(size_t)(k + kb + 1) * HD + col0 + m];
        c2 = __builtin_amdgcn_wmma_f32_16x16x4_f32(false, a, false, b,
                                                   (short)0, c2, false, false);
    }
    {
        const float bias2 = b2[col0 + m];
#pragma unroll
        for (int r = 0; r < 8; ++r) {
            int row = row0 + r + 8 * half;
            if (row < N)
                h_out[(size_t)row * HD + col0 + m] = c2[r] + bias2;
        }
    }
}

// ---------------------------------------------------------------------------
// prop: out[dst] += (-dis[src]*dis[dst]) * x[src]   (out pre-zeroed)
// One wave per edge; lane covers features {2*lane, 2*lane+1} via one b64 load.
// Gathers and atomic destination are L2-resident (12.8 MB each vs 192 MB L2).
// ---------------------------------------------------------------------------
__global__ __launch_bounds__(256) void prop_kernel(const float* __restrict__ x,
                                                   const int* __restrict__ src,
                                                   const int* __restrict__ dst,
                                                   const float* __restrict__ dis,
                                                   float* __restrict__ out, int E) {
    int e    = blockIdx.x * 8 + (threadIdx.x >> 5);
    int lane = threadIdx.x & 31;
    if (e >= E) return;
    int s = src[e];
    int d = dst[e];
    float w = -dis[s] * dis[d];
    const float2 v = *(const float2*)(x + (size_t)s * HD + 2 * lane);
    float* od = out + (size_t)d * HD + 2 * lane;
    atomicAdd(od,     w * v.x);
    atomicAdd(od + 1, w * v.y);
}

// out = 0.5*coe[0]*t0 + coe[1]*t1      (float4 per thread; n4 = N*HD/4)
__global__ __launch_bounds__(256) void init_out_kernel(const float4* __restrict__ t0,
                                                       const float4* __restrict__ t1,
                                                       const float* __restrict__ coe,
                                                       float4* __restrict__ out, size_t n4) {
    size_t i = (size_t)blockIdx.x * blockDim.x + threadIdx.x;
    if (i < n4) {
        float c0 = 0.5f * coe[0], c1 = coe[1];
        float4 a = t0[i], b = t1[i], r;
        r.x = c0 * a.x + c1 * b.x;
        r.y = c0 * a.y + c1 * b.y;
        r.z = c0 * a.z + c1 * b.z;
        r.w = c0 * a.w + c1 * b.w;
        out[i] = r;
    }
}

// t2 = 2*p - t0 (in place on p); out += coe[ci]*t2   (float4 per thread)
__global__ __launch_bounds__(256) void cheb_step_kernel(float4* __restrict__ p,
                                                        const float4* __restrict__ t0,
                                                        const float* __restrict__ coe,
                                                        int ci,
                                                        float4* __restrict__ out, size_t n4) {
    size_t i = (size_t)blockIdx.x * blockDim.x + threadIdx.x;
    if (i < n4) {
        float cc = coe[ci];
        float4 pv = p[i], a = t0[i], o = out[i], t;
        t.x = 2.0f * pv.x - a.x;
        t.y = 2.0f * pv.y - a.y;
        t.z = 2.0f * pv.z - a.z;
        t.w = 2.0f * pv.w - a.w;
        p[i] = t;
        o.x += cc * t.x; o.y += cc * t.y; o.z += cc * t.z; o.w += cc * t.w;
        out[i] = o;
    }
}

// ---------------------------------------------------------------------------
extern "C" void kernel_launch(void* const* d_in, const int* in_sizes, int n_in,
                              void* d_out, int out_size, void* d_ws, size_t ws_size,
                              hipStream_t stream) {
    const float* X    = (const float*)d_in[0];
    const int*   ei   = (const int*)d_in[1];
    const float* W1   = (const float*)d_in[2];
    const float* b1   = (const float*)d_in[3];
    const float* W2   = (const float*)d_in[4];
    const float* b2   = (const float*)d_in[5];
    const float* temp = (const float*)d_in[6];

    const int N = in_sizes[0] / F_IN;
    const int E = in_sizes[1] / 2;
    const int* src = ei;         // edge_index[0]
    const int* dst = ei + E;     // edge_index[1]

    const size_t NH = (size_t)N * HD;
    const size_t N4 = NH / 4;    // HD=64 -> always divisible

    // Workspace layout
    float* h   = (float*)d_ws;   // Tx0 initially (MLP output)
    float* t1  = h + NH;
    float* t2  = t1 + NH;
    float* deg = t2 + NH;
    float* dis = deg + N;
    float* coe = dis + N;        // 7 floats
    float* out = (float*)d_out;

    // Degree + normalization
    hipMemsetAsync(deg, 0, (size_t)N * sizeof(float), stream);
    coe_kernel<<<1, 1, 0, stream>>>(temp, coe);
    deg_kernel<<<(E + 255) / 256, 256, 0, stream>>>(src, deg, E);
    dis_kernel<<<(N + 255) / 256, 256, 0, stream>>>(deg, dis, N);

    // Fused MLP (WMMA f32 path)
    mlp_kernel<<<(N + 15) / 16, 128, 0, stream>>>(X, W1, b1, W2, b2, h, N);

    // Tx1 = prop(Tx0)
    hipMemsetAsync(t1, 0, NH * sizeof(float), stream);
    prop_kernel<<<(E + 7) / 8, 256, 0, stream>>>(h, src, dst, dis, t1, E);

    // out = coe0/2 * Tx0 + coe1 * Tx1
    init_out_kernel<<<(int)((N4 + 255) / 256), 256, 0, stream>>>(
        (const float4*)h, (const float4*)t1, coe, (float4*)out, N4);

    // Chebyshev recurrence: Tx2 = 2*prop(Tx1) - Tx0 ; out += coe_i * Tx2
    float* T0 = h;
    float* T1 = t1;
    float* T2 = t2;
    for (int i = 2; i <= KCH; ++i) {
        hipMemsetAsync(T2, 0, NH * sizeof(float), stream);
        prop_kernel<<<(E + 7) / 8, 256, 0, stream>>>(T1, src, dst, dis, T2, E);
        cheb_step_kernel<<<(int)((N4 + 255) / 256), 256, 0, stream>>>(
            (float4*)T2, (const float4*)T0, coe, i, (float4*)out, N4);
        float* tmp = T0; T0 = T1; T1 = T2; T2 = tmp;
    }
}